// GraphEmbedding_72052371357806
// MI455X (gfx1250) — compile-verified
//
#include <hip/hip_runtime.h>
#include <math.h>
#include <stdint.h>

typedef float v2f __attribute__((ext_vector_type(2)));
typedef float v8f __attribute__((ext_vector_type(8)));

// ---------------- problem constants ----------------
#define BSZ   512
#define NNB   20
#define DIM   172
#define TDIM  100
#define GDIM  50
#define MROWS 10752          /* 512 + 512*20 */
#define MTILES 672           /* MROWS/16 */

// padded MLP dims (multiples of 16 / 4)
#define K1 64    /* t0 padded 50->64   */
#define N1 112   /* h1 padded 100->112 */
#define N2 208   /* h2 padded 200->208 */
#define NF 400   /* fourier feat dim   */
#define N3 208   /* h3 padded 200->208 */
#define N4 400   /* final S dim        */

// output offsets (floats)
#define OFF_SRC   0u
#define OFF_NEIGH 292864u
#define OFF_STIME 6150144u
#define OFF_ETIME 6201344u
#define OFF_EDGEF 7225344u
#define OFF_MASK  8986624u

__device__ __forceinline__ float gelu_tanh(float x) {
  const float k0 = 0.7978845608028654f; // sqrt(2/pi)
  return 0.5f * x * (1.0f + tanhf(k0 * (x + 0.044715f * x * x * x)));
}

// ---------------- pad/copy a 2D weight into zero-padded layout ----------------
__global__ void k_pad2d(const float* __restrict__ src, float* __restrict__ dst,
                        int sr, int sc, int dr, int dc) {
  int i = blockIdx.x * blockDim.x + threadIdx.x;
  int total = dr * dc;
  if (i >= total) return;
  int r = i / dc, c = i % dc;
  dst[i] = (r < sr && c < sc) ? src[r * sc + c] : 0.0f;
}

// ---------------- gathers / elementwise outputs ----------------
__global__ void k_src_feat(const int* __restrict__ src, const float* __restrict__ nf,
                           const float* __restrict__ mem, float* __restrict__ out) {
  int i = blockIdx.x * blockDim.x + threadIdx.x;
  if (i >= BSZ * DIM) return;
  int b = i / DIM, d = i % DIM;
  size_t n = (size_t)src[b] * DIM + d;
  out[OFF_SRC + (size_t)b * 572 + d] = nf[n] + mem[n];
}

__global__ void k_neigh_feat(const int* __restrict__ nbr, const float* __restrict__ nf,
                             const float* __restrict__ mem, float* __restrict__ out) {
  int i = blockIdx.x * blockDim.x + threadIdx.x;
  if (i >= BSZ * NNB * DIM) return;
  int j = i / DIM, d = i % DIM;
  size_t n = (size_t)nbr[j] * DIM + d;
  out[OFF_NEIGH + (size_t)j * 572 + d] = nf[n] + mem[n];
}

__global__ void k_edge_feat(const int* __restrict__ eidx, const float* __restrict__ ef,
                            float* __restrict__ out) {
  int i = blockIdx.x * blockDim.x + threadIdx.x;
  if (i >= BSZ * NNB * DIM) return;
  int j = i / DIM, d = i % DIM;
  out[OFF_EDGEF + i] = ef[(size_t)eidx[j] * DIM + d];
}

__global__ void k_src_time(const float* __restrict__ tb, float* __restrict__ out) {
  int i = blockIdx.x * blockDim.x + threadIdx.x;
  if (i >= BSZ * TDIM) return;
  out[OFF_STIME + i] = cosf(tb[i % TDIM]);   // cos(0*w + b)
}

__global__ void k_edge_time(const float* __restrict__ ts, const float* __restrict__ et,
                            const float* __restrict__ tw, const float* __restrict__ tb,
                            float* __restrict__ out) {
  int i = blockIdx.x * blockDim.x + threadIdx.x;
  if (i >= BSZ * NNB * TDIM) return;
  int j = i / TDIM, t = i % TDIM;
  float delta = ts[j / NNB] - et[j];
  out[OFF_ETIME + i] = cosf(delta * tw[t] + tb[t]);
}

__global__ void k_mask(const int* __restrict__ nbr, float* __restrict__ out) {
  int i = blockIdx.x * blockDim.x + threadIdx.x;
  if (i >= BSZ * NNB) return;
  out[OFF_MASK + i] = (nbr[i] == 0) ? 1.0f : 0.0f;
}

// ---------------- gene reduction: t0 = ((colsum gene) @ Wm + G*bm) * tm0 ----------------
__global__ __launch_bounds__(256)
void k_gene_reduce(const int* __restrict__ src, const int* __restrict__ nbr,
                   const float* __restrict__ gene, const float* __restrict__ Wm,
                   const float* __restrict__ bm, float* __restrict__ t0) {
  int inst = blockIdx.x;
  int node = (inst < BSZ) ? src[inst] : nbr[inst - BSZ];
  const float* g = gene + (size_t)node * (GDIM * DIM);
  __shared__ float s[DIM];
  __shared__ float mflag[64];
  __shared__ float tm0s;
  int tid = threadIdx.x;

  // column sums (coalesced across d for each g)
  for (int d = tid; d < DIM; d += 256) {
    float acc = 0.0f;
    for (int r = 0; r < GDIM; ++r) acc += g[r * DIM + d];
    s[d] = acc;
  }
  // per-row sign gate
  if (tid < 64) mflag[tid] = 0.0f;
  if (tid < GDIM) {
    float mx = -INFINITY, mn = INFINITY;
    for (int d = 0; d < DIM; ++d) {
      float v = g[tid * DIM + d];
      mx = fmaxf(mx, v); mn = fminf(mn, v);
    }
    mflag[tid] = (fabsf(mx) + fabsf(mn)) > 0.0f ? 1.0f : 0.0f;
  }
  __syncthreads();
  if (tid == 0) {
    float sum = 0.0f;
    for (int r = 0; r < GDIM; ++r) sum += mflag[r];
    tm0s = (sum > 0.0f) ? 1.0f : 0.0f;
  }
  __syncthreads();
  if (tid < K1) {
    float v = 0.0f;
    if (tid < GDIM) {
      float acc = 50.0f * bm[tid];
      for (int d = 0; d < DIM; ++d) acc += s[d] * Wm[d * GDIM + tid];
      v = acc * tm0s;
    }
    t0[(size_t)inst * K1 + tid] = v;
  }
}

// ---------------- fp32 WMMA GEMM: out = epilogue(A[M,K] @ Bw[K,ldb] + bias) ----------------
// grid.x = m-tile, 4 waves per block share one LDS A-tile (streamed in via the
// CDNA5 async global->LDS path, ASYNCcnt-tracked) and split N-tiles.
__global__ __launch_bounds__(128)
void k_gemm_wmma(const float* __restrict__ A, int K,
                 const float* __restrict__ Bw, int ldb,
                 const float* __restrict__ bias,
                 float* __restrict__ out0, float* __restrict__ out1,
                 int Ncols, int ldo, int mode) {
  __shared__ float As[16 * 400];
  int m0 = blockIdx.x * 16;
  const float* Ablk = A + (size_t)m0 * K;          // lda == K for all stages

  // ---- async A-tile stage: global -> LDS, bypassing VGPRs ----
  // low 32 bits of a generic LDS pointer == LDS byte offset
  unsigned ldsBase = (unsigned)(uintptr_t)(void*)As;
  int nVec = (16 * K) >> 2;                        // number of 16B transfers
  for (int v = threadIdx.x; v < nVec; v += 128) {
    unsigned lds = ldsBase + ((unsigned)v << 4);
    const float* gp = Ablk + ((size_t)v << 2);
    asm volatile("global_load_async_to_lds_b128 %0, %1, off"
                 :: "v"(lds), "v"(gp) : "memory");
  }
  asm volatile("s_wait_asynccnt 0" ::: "memory");
  __syncthreads();

  int lane = threadIdx.x & 31;
  int wave = threadIdx.x >> 5;
  int half = lane >> 4;      // 0: K+{0,1}/rows 0..7   1: K+{2,3}/rows 8..15
  int l16  = lane & 15;

  for (int nt = wave * 16; nt < Ncols; nt += 64) {
    int col = nt + l16;
    __builtin_prefetch(&Bw[col], 0, 1);            // global_prefetch_b8
    v8f acc = {0.f, 0.f, 0.f, 0.f, 0.f, 0.f, 0.f, 0.f};
    for (int k = 0; k < K; k += 4) {
      int ka = k + 2 * half;
      v2f a, b;
      a.x = As[l16 * K + ka];
      a.y = As[l16 * K + ka + 1];
      b.x = Bw[(size_t)ka * ldb + col];
      b.y = Bw[(size_t)(ka + 1) * ldb + col];
      acc = __builtin_amdgcn_wmma_f32_16x16x4_f32(
          /*neg_a=*/false, a, /*neg_b=*/false, b,
          /*c_mod=*/(short)0, acc, /*reuse_a=*/false, /*reuse_b=*/false);
    }
    float bv = bias ? bias[col] : 0.0f;
    int rbase = half * 8;
    for (int r = 0; r < 8; ++r) {
      int m = m0 + rbase + r;
      float v = acc[r];
      if (mode == 0) {                 // h1 = gelu(x + b)
        out0[(size_t)m * ldo + col] = gelu_tanh(v + bv);
      } else if (mode == 1) {          // F = 0.05*[cos(h), sin(h)]
        float h = v + bv;
        if (col < 200) {
          out0[(size_t)m * NF + col]       = 0.05f * cosf(h);
          out0[(size_t)m * NF + col + 200] = 0.05f * sinf(h);
        }
      } else if (mode == 2) {          // h3 = gelu(x)
        out0[(size_t)m * ldo + col] = gelu_tanh(v);
      } else {                         // final S: scatter into src_cat / neigh_cat
        float* dst = (m < BSZ) ? (out0 + (size_t)m * 572 + DIM)
                               : (out1 + (size_t)(m - BSZ) * 572 + DIM);
        dst[col] = v;
      }
    }
  }
}

// ---------------- launch ----------------
extern "C" void kernel_launch(void* const* d_in, const int* in_sizes, int n_in,
                              void* d_out, int out_size, void* d_ws, size_t ws_size,
                              hipStream_t stream) {
  const int*   src  = (const int*)d_in[0];
  const float* ts   = (const float*)d_in[1];
  const int*   nbr  = (const int*)d_in[2];
  const int*   eidx = (const int*)d_in[3];
  const float* et   = (const float*)d_in[4];
  const float* nf   = (const float*)d_in[5];
  const float* gene = (const float*)d_in[6];
  const float* ef   = (const float*)d_in[7];
  const float* mem  = (const float*)d_in[8];
  const float* tw   = (const float*)d_in[9];
  const float* tb   = (const float*)d_in[10];
  const float* Wm   = (const float*)d_in[11];
  const float* bm   = (const float*)d_in[12];
  const float* W1   = (const float*)d_in[13];
  const float* b1   = (const float*)d_in[14];
  const float* W2   = (const float*)d_in[15];
  const float* b2   = (const float*)d_in[16];
  const float* P1   = (const float*)d_in[17];
  const float* P2   = (const float*)d_in[18];
  float* out = (float*)d_out;
  float* ws  = (float*)d_ws;

  // workspace layout (floats)
  size_t o = 0;
  float* t0  = ws + o; o += (size_t)MROWS * K1;
  float* h1  = ws + o; o += (size_t)MROWS * N1;
  float* fbf = ws + o; o += (size_t)MROWS * NF;
  float* h3  = ws + o; o += (size_t)MROWS * N3;
  float* W1p = ws + o; o += (size_t)K1 * N1;
  float* b1p = ws + o; o += N1;
  float* W2p = ws + o; o += (size_t)N1 * N2;
  float* b2p = ws + o; o += N2;
  float* P1p = ws + o; o += (size_t)NF * N3;
  float* P2p = ws + o; o += (size_t)N3 * N4;

  const int T = 256;
  auto cdiv = [](int a, int b) { return (a + b - 1) / b; };

  // pack zero-padded weights
  k_pad2d<<<cdiv(K1 * N1, T), T, 0, stream>>>(W1, W1p, GDIM, 100, K1, N1);
  k_pad2d<<<cdiv(N1, T),      T, 0, stream>>>(b1, b1p, 1, 100, 1, N1);
  k_pad2d<<<cdiv(N1 * N2, T), T, 0, stream>>>(W2, W2p, 100, 200, N1, N2);
  k_pad2d<<<cdiv(N2, T),      T, 0, stream>>>(b2, b2p, 1, 200, 1, N2);
  k_pad2d<<<cdiv(NF * N3, T), T, 0, stream>>>(P1, P1p, NF, 200, NF, N3);
  k_pad2d<<<cdiv(N3 * N4, T), T, 0, stream>>>(P2, P2p, 200, N4, N3, N4);

  // gathers / elementwise outputs
  k_src_feat  <<<cdiv(BSZ * DIM, T),        T, 0, stream>>>(src, nf, mem, out);
  k_neigh_feat<<<cdiv(BSZ * NNB * DIM, T),  T, 0, stream>>>(nbr, nf, mem, out);
  k_edge_feat <<<cdiv(BSZ * NNB * DIM, T),  T, 0, stream>>>(eidx, ef, out);
  k_src_time  <<<cdiv(BSZ * TDIM, T),       T, 0, stream>>>(tb, out);
  k_edge_time <<<cdiv(BSZ * NNB * TDIM, T), T, 0, stream>>>(ts, et, tw, tb, out);
  k_mask      <<<cdiv(BSZ * NNB, T),        T, 0, stream>>>(nbr, out);

  // gene reduction -> t0 [M, 64]
  k_gene_reduce<<<MROWS, 256, 0, stream>>>(src, nbr, gene, Wm, bm, t0);

  // batched MLP via fp32 WMMA
  k_gemm_wmma<<<MTILES, 128, 0, stream>>>(t0,  K1, W1p, N1, b1p, h1,  nullptr, N1, N1, 0);
  k_gemm_wmma<<<MTILES, 128, 0, stream>>>(h1,  N1, W2p, N2, b2p, fbf, nullptr, N2, 0,  1);
  k_gemm_wmma<<<MTILES, 128, 0, stream>>>(fbf, NF, P1p, N3, nullptr, h3, nullptr, N3, N3, 2);
  k_gemm_wmma<<<MTILES, 128, 0, stream>>>(h3,  N3, P2p, N4, nullptr,
                                          out + OFF_SRC, out + OFF_NEIGH, N4, 0, 3);
}